// VariableClementsDiag1_6047313953011
// MI455X (gfx1250) — compile-verified
//
#include <hip/hip_runtime.h>
#include <math.h>

// ---------------------------------------------------------------------------
// VariableClementsDiag1 on MI455X (gfx1250)
//
// forward() is complex-linear: out = M @ (xr + i*xi) with M = forward(I),
// M is 256x256 complex (512 KB as two f32 planes -> L2 resident).
// Step 1: build M by simulating the mesh on identity columns.  128-layer
//         serial chain -> minimize per-layer latency: 2 columns per block
//         (128 blocks), one pair-slot per thread per layer, 3 barriers/step.
// Step 2: complex GEMM 256x256 x 256x2048 via V_WMMA_F32_16X16X4_F32.
//         Each wave: 4 stacked 16x16 M-tiles for one 16-column stripe.
//         Two alternating fragment buffers (no rotating copy) keep loads in
//         flight under each 16-WMMA block with a single partial wait.
// ---------------------------------------------------------------------------

typedef float v2f __attribute__((ext_vector_type(2)));
typedef float v8f __attribute__((ext_vector_type(8)));

#define NM       256     // modes (rows)
#define NSTEPS   64      // scan steps (each = layerA + layerB)
#define HALF     128     // pairs in layerA
#define HALFB    127     // pairs in layerB
#define TC       2       // identity columns per workgroup in the mesh build
#define INV_SQRT2 0.70710678118654752440f

__device__ __forceinline__ void psdc2(float& xr, float& xi, float& yr, float& yi,
                                      float e0r, float e0i, float e1r, float e1i)
{
    // psdc with e0
    float ixr = e0r * xr - e0i * xi;
    float ixi = e0r * xi + e0i * xr;
    float nxr = INV_SQRT2 * (ixr - yi), nxi = INV_SQRT2 * (ixi + yr);
    float nyr = INV_SQRT2 * (yr - ixi), nyi = INV_SQRT2 * (yi + ixr);
    // psdc with e1
    ixr = e1r * nxr - e1i * nxi;
    ixi = e1r * nxi + e1i * nxr;
    xr = INV_SQRT2 * (ixr - nyi);  xi = INV_SQRT2 * (ixi + nyr);
    yr = INV_SQRT2 * (nyr - ixi);  yi = INV_SQRT2 * (nyi + ixr);
}

// ---------------------------------------------------------------------------
// Kernel 1: build W = forward(Identity).  Grid: NM/TC = 128 blocks x 256 thr.
// LDS: cr/ci [NM][2] (4 KB) + trigA (2 KB) + trigB (2 KB).
// ---------------------------------------------------------------------------
__global__ __launch_bounds__(256)
void mesh_build_w(const float* __restrict__ A0, const float* __restrict__ A1,
                  const float* __restrict__ B0, const float* __restrict__ B1,
                  const float* __restrict__ diag,
                  float* __restrict__ Wr, float* __restrict__ Wi)
{
    __shared__ float cr[NM * TC];
    __shared__ float ci[NM * TC];
    __shared__ float trigA[HALF * 4];    // e0r,e0i,e1r,e1i per layer-A pair
    __shared__ float trigB[HALFB * 4];   // e0r,e0i,e1r,e1i per layer-B pair

    const int tid     = threadIdx.x;
    const int colBase = blockIdx.x * TC;

    // identity columns: cr[r][j] = (r == colBase + j)
    for (int s = tid; s < NM * TC; s += 256) {
        const int r = s >> 1, j = s & 1;
        cr[s] = (r == colBase + j) ? 1.0f : 0.0f;
        ci[s] = 0.0f;
    }
    __syncthreads();

    for (int step = 0; step < NSTEPS; ++step) {
        // ---- fill both trig tables for this step (2 sincos per thread) ----
        if (tid < HALF) {
            float s0, c0, s1, c1;
            __sincosf(A0[step * HALF + tid], &s0, &c0);
            __sincosf(A1[step * HALF + tid], &s1, &c1);
            trigA[tid * 4 + 0] = c0;  trigA[tid * 4 + 1] = s0;
            trigA[tid * 4 + 2] = c1;  trigA[tid * 4 + 3] = s1;
        } else if (tid - HALF < HALFB) {
            const int p = tid - HALF;
            float s0, c0, s1, c1;
            __sincosf(B0[step * HALFB + p], &s0, &c0);
            __sincosf(B1[step * HALFB + p], &s1, &c1);
            trigB[p * 4 + 0] = c0;  trigB[p * 4 + 1] = s0;
            trigB[p * 4 + 2] = c1;  trigB[p * 4 + 3] = s1;
        }
        __syncthreads();

        // ---- layer A: pairs (2p, 2p+1); exactly one slot per thread ----
        {
            const int p = tid >> 1, j = tid & 1;
            const int i0 = (2 * p) * TC + j, i1 = i0 + TC;
            float xr = cr[i0], xi = ci[i0], yr = cr[i1], yi = ci[i1];
            psdc2(xr, xi, yr, yi,
                  trigA[p * 4 + 0], trigA[p * 4 + 1],
                  trigA[p * 4 + 2], trigA[p * 4 + 3]);
            cr[i0] = xr; ci[i0] = xi; cr[i1] = yr; ci[i1] = yi;
        }
        __syncthreads();

        // ---- layer B: pairs (2p+1, 2p+2); one (guarded) slot per thread ----
        if (tid < HALFB * TC) {
            const int p = tid >> 1, j = tid & 1;
            const int i0 = (2 * p + 1) * TC + j, i1 = i0 + TC;
            float xr = cr[i0], xi = ci[i0], yr = cr[i1], yi = ci[i1];
            psdc2(xr, xi, yr, yi,
                  trigB[p * 4 + 0], trigB[p * 4 + 1],
                  trigB[p * 4 + 2], trigB[p * 4 + 3]);
            cr[i0] = xr; ci[i0] = xi; cr[i1] = yr; ci[i1] = yi;
        }
        __syncthreads();
    }

    // diagonal phase + writeback of the W planes
    for (int s = tid; s < NM * TC; s += 256) {
        const int r = s >> 1, j = s & 1;
        float sn, cs;
        __sincosf(diag[r], &sn, &cs);
        const float vr = cr[s], vi = ci[s];
        Wr[r * NM + colBase + j] = cs * vr - sn * vi;
        Wi[r * NM + colBase + j] = cs * vi + sn * vr;
    }
}

// ---------------------------------------------------------------------------
// Kernel 2: out = W @ (xr + i*xi) via fp32 WMMA 16x16x4, software-pipelined
// with two alternating fragment buffers (no rotating copy).
// A (16x4 f32) layout: lane<16 -> row=lane,   k = k0+{0,1}
//                      lane>=16-> row=lane-16, k = k0+{2,3}
// B (4x16 f32) layout: v0 holds rows k0+(lane>>4), v1 rows k0+2+(lane>>4),
//                      col = tn*16 + (lane&15).
// D (16x16 f32): acc[v] is element (m = v + 8*(lane>>4), n = lane&15).
// ---------------------------------------------------------------------------
struct Frags {
    v2f Ar0, Ai0, Ar1, Ai1, Ar2, Ai2, Ar3, Ai3;   // 4 M-tiles, re/im planes
    v2f Br, Bi;                                    // shared B fragment
};

__device__ __forceinline__ Frags load_frags(const float* __restrict__ wrRow,
                                            const float* __restrict__ wiRow,
                                            const float* __restrict__ xr,
                                            const float* __restrict__ xi,
                                            int k0, int hi, int boff, int Bcols)
{
    Frags f;
    const int ak = k0 + 2 * hi;                    // contiguous pair {ak, ak+1}
    f.Ar0 = *reinterpret_cast<const v2f*>(wrRow + ak);
    f.Ai0 = *reinterpret_cast<const v2f*>(wiRow + ak);
    f.Ar1 = *reinterpret_cast<const v2f*>(wrRow + 16 * NM + ak);
    f.Ai1 = *reinterpret_cast<const v2f*>(wiRow + 16 * NM + ak);
    f.Ar2 = *reinterpret_cast<const v2f*>(wrRow + 32 * NM + ak);
    f.Ai2 = *reinterpret_cast<const v2f*>(wiRow + 32 * NM + ak);
    f.Ar3 = *reinterpret_cast<const v2f*>(wrRow + 48 * NM + ak);
    f.Ai3 = *reinterpret_cast<const v2f*>(wiRow + 48 * NM + ak);

    const float* pr = xr + (size_t)k0 * Bcols;     // scalar base (GVS form)
    const float* pi = xi + (size_t)k0 * Bcols;
    f.Br.x = pr[boff];
    f.Br.y = pr[boff + 2 * Bcols];
    f.Bi.x = pi[boff];
    f.Bi.y = pi[boff + 2 * Bcols];
    return f;
}

#define WMMA16(f)                                                                                        \
    accRR[0] = __builtin_amdgcn_wmma_f32_16x16x4_f32(false, (f).Ar0, false, (f).Br, (short)0, accRR[0], false, false); \
    accRI[0] = __builtin_amdgcn_wmma_f32_16x16x4_f32(false, (f).Ar0, false, (f).Bi, (short)0, accRI[0], false, false); \
    accII[0] = __builtin_amdgcn_wmma_f32_16x16x4_f32(false, (f).Ai0, false, (f).Bi, (short)0, accII[0], false, false); \
    accIR[0] = __builtin_amdgcn_wmma_f32_16x16x4_f32(false, (f).Ai0, false, (f).Br, (short)0, accIR[0], false, false); \
    accRR[1] = __builtin_amdgcn_wmma_f32_16x16x4_f32(false, (f).Ar1, false, (f).Br, (short)0, accRR[1], false, false); \
    accRI[1] = __builtin_amdgcn_wmma_f32_16x16x4_f32(false, (f).Ar1, false, (f).Bi, (short)0, accRI[1], false, false); \
    accII[1] = __builtin_amdgcn_wmma_f32_16x16x4_f32(false, (f).Ai1, false, (f).Bi, (short)0, accII[1], false, false); \
    accIR[1] = __builtin_amdgcn_wmma_f32_16x16x4_f32(false, (f).Ai1, false, (f).Br, (short)0, accIR[1], false, false); \
    accRR[2] = __builtin_amdgcn_wmma_f32_16x16x4_f32(false, (f).Ar2, false, (f).Br, (short)0, accRR[2], false, false); \
    accRI[2] = __builtin_amdgcn_wmma_f32_16x16x4_f32(false, (f).Ar2, false, (f).Bi, (short)0, accRI[2], false, false); \
    accII[2] = __builtin_amdgcn_wmma_f32_16x16x4_f32(false, (f).Ai2, false, (f).Bi, (short)0, accII[2], false, false); \
    accIR[2] = __builtin_amdgcn_wmma_f32_16x16x4_f32(false, (f).Ai2, false, (f).Br, (short)0, accIR[2], false, false); \
    accRR[3] = __builtin_amdgcn_wmma_f32_16x16x4_f32(false, (f).Ar3, false, (f).Br, (short)0, accRR[3], false, false); \
    accRI[3] = __builtin_amdgcn_wmma_f32_16x16x4_f32(false, (f).Ar3, false, (f).Bi, (short)0, accRI[3], false, false); \
    accII[3] = __builtin_amdgcn_wmma_f32_16x16x4_f32(false, (f).Ai3, false, (f).Bi, (short)0, accII[3], false, false); \
    accIR[3] = __builtin_amdgcn_wmma_f32_16x16x4_f32(false, (f).Ai3, false, (f).Br, (short)0, accIR[3], false, false)

__global__ __launch_bounds__(128)
void cgemm_wmma(const float* __restrict__ Wr, const float* __restrict__ Wi,
                const float* __restrict__ xr, const float* __restrict__ xi,
                float* __restrict__ out, int Bcols, int nJobs)
{
    const int lane = threadIdx.x & 31;
    const int wave = threadIdx.x >> 5;
    const int job  = blockIdx.x * 4 + wave;     // (M-group, tn) pair
    if (job >= nJobs) return;

    const int mGroups = NM / 64;                // 4 groups of 4 tiles
    const int tg = job % mGroups;               // M group (64 rows)
    const int tn = job / mGroups;               // 16-column stripe

    const int hi  = lane >> 4;                  // 0 | 1
    const int l15 = lane & 15;
    const int am  = tg * 64 + l15;              // first W row for A fragments
    const int bn  = tn * 16 + l15;              // x column for B fragment
    const int boff = hi * Bcols + bn;           // per-lane 32-bit B offset

    v8f accRR[4] = {{}, {}, {}, {}};
    v8f accII[4] = {{}, {}, {}, {}};
    v8f accRI[4] = {{}, {}, {}, {}};
    v8f accIR[4] = {{}, {}, {}, {}};

    const float* wrRow = Wr + (size_t)am * NM;  // rows am, am+16, am+32, am+48
    const float* wiRow = Wi + (size_t)am * NM;

    // ---- two-stage software pipeline, alternating buffers, no copies:
    //      each buffer is reloaded in place while the other's WMMAs run.
    Frags f0 = load_frags(wrRow, wiRow, xr, xi, 0, hi, boff, Bcols);
    Frags f1 = load_frags(wrRow, wiRow, xr, xi, 4, hi, boff, Bcols);

#pragma unroll 1
    for (int k0 = 8; k0 < NM; k0 += 8) {        // 31 iterations, 2 k-steps each
        WMMA16(f0);
        f0 = load_frags(wrRow, wiRow, xr, xi, k0, hi, boff, Bcols);
        WMMA16(f1);
        f1 = load_frags(wrRow, wiRow, xr, xi, k0 + 4, hi, boff, Bcols);
    }
    // epilogue: last two k-steps
    WMMA16(f0);
    WMMA16(f1);

    // combine to complex64 (interleaved re,im) and store 8B per lane
#pragma unroll
    for (int t = 0; t < 4; ++t) {
#pragma unroll
        for (int v = 0; v < 8; ++v) {
            const int m = tg * 64 + t * 16 + v + 8 * hi;
            const int n = tn * 16 + l15;
            float2 val;
            val.x = accRR[t][v] - accII[t][v];
            val.y = accRI[t][v] + accIR[t][v];
            *reinterpret_cast<float2*>(out + ((size_t)m * Bcols + n) * 2) = val;
        }
    }
}

// ---------------------------------------------------------------------------
extern "C" void kernel_launch(void* const* d_in, const int* in_sizes, int n_in,
                              void* d_out, int out_size, void* d_ws, size_t ws_size,
                              hipStream_t stream)
{
    const float* xr   = (const float*)d_in[0];
    const float* xi   = (const float*)d_in[1];
    const float* A0   = (const float*)d_in[2];
    const float* A1   = (const float*)d_in[3];
    const float* B0   = (const float*)d_in[4];
    const float* B1   = (const float*)d_in[5];
    const float* diag = (const float*)d_in[6];
    float* out = (float*)d_out;

    const int Bcols = in_sizes[0] / NM;            // 2048

    float* Wr = (float*)d_ws;                      // 256x256 f32
    float* Wi = Wr + NM * NM;                      // 256x256 f32 (512 KB total)

    // Step 1: compose the mesh into W (128 workgroups of 2 identity columns;
    //         one pair-slot per thread per layer -> minimal serial latency)
    mesh_build_w<<<NM / TC, 256, 0, stream>>>(A0, A1, B0, B1, diag, Wr, Wi);

    // Step 2: complex GEMM out = W @ x on the WMMA units.
    //         jobs = (M groups of 64 rows) x (16-column stripes)
    const int nJobs  = (NM / 64) * (Bcols / 16);   // 4 * 128 = 512 waves
    const int nBlocks = (nJobs + 3) / 4;           // 4 waves per block
    cgemm_wmma<<<nBlocks, 128, 0, stream>>>(Wr, Wi, xr, xi, out, Bcols, nJobs);
}